// Decoder_2284922602102
// MI455X (gfx1250) — compile-verified
//
#include <hip/hip_runtime.h>
#include <hip/hip_bf16.h>

typedef __attribute__((ext_vector_type(2))) float v2f;
typedef __attribute__((ext_vector_type(8))) float v8f;

#define B_TOT 4096
#define T_LEN 256
#define IN_D  16
#define H1_D  64
#define LDH   68   // padded LDS row stride (floats): bank = (4m+k)%64, conflict-free

__device__ __forceinline__ float sigf(float x) {
    return 1.0f / (1.0f + __expf(-x));
}

// One workgroup = 16 batch rows through all 256 timesteps, both LSTM layers + FC fused.
// 4 waves; wave w owns h-slice [16w,16w+16) i.e. gate tiles {w, w+4, w+8, w+12}.
__global__ __launch_bounds__(128)
void lstm2_fused_kernel(const float* __restrict__ x,
                        const float* __restrict__ w_ih1, const float* __restrict__ w_hh1,
                        const float* __restrict__ b_ih1, const float* __restrict__ b_hh1,
                        const float* __restrict__ w_ih2, const float* __restrict__ w_hh2,
                        const float* __restrict__ b_ih2, const float* __restrict__ b_hh2,
                        const float* __restrict__ fc_w,  const float* __restrict__ fc_b,
                        float* __restrict__ out)
{
    __shared__ float hbuf[2][16 * LDH];   // double-buffered h1 state [m][hidx]
    __shared__ float g2buf[16 * 4];       // layer-2 gate scratch (wave 0 only)

    const int tid   = threadIdx.x;
    const int wave  = tid >> 5;
    const int lane  = tid & 31;
    const int lhalf = lane >> 4;          // 0: lanes 0-15, 1: lanes 16-31
    const int l15   = lane & 15;
    const int k0    = lhalf * 2;          // K offset within a K=4 chunk (ISA A/B layout)
    const int b0    = blockIdx.x * 16;

    // zero initial h state (h0 = 0); buffer 0 is read at t=0
    for (int i = tid; i < 2 * 16 * LDH; i += 128) (&hbuf[0][0])[i] = 0.f;

    // ---------- layer-1 weight fragments, register-resident for all 256 steps ----------
    // B-frag layout (16x16x4 f32): lane l holds B[k0][n], B[k0+1][n] with n = l&15
    v2f Bh[4][16];      // W_hh1^T  (K=64 -> 16 chunks)  [gate][kchunk]
    v2f Bx[4][4];       // W_ih1^T  (K=16 -> 4 chunks)
    float bias1[4];
    #pragma unroll
    for (int gi = 0; gi < 4; ++gi) {
        const int n = gi * 64 + wave * 16 + l15;          // global gate row 0..255
        const float* wh = w_hh1 + n * H1_D + k0;
        #pragma unroll
        for (int kk = 0; kk < 16; ++kk) Bh[gi][kk] = *(const v2f*)(wh + 4 * kk);
        const float* wx = w_ih1 + n * IN_D + k0;
        #pragma unroll
        for (int kk = 0; kk < 4; ++kk)  Bx[gi][kk] = *(const v2f*)(wx + 4 * kk);
        bias1[gi] = b_ih1[n] + b_hh1[n];
    }

    // ---------- layer-2 fragments (used by wave 0; N padded 4 -> 16 with zeros) ----------
    v2f B2[16];
    float bias2 = 0.f;
    #pragma unroll
    for (int kk = 0; kk < 16; ++kk) { B2[kk].x = 0.f; B2[kk].y = 0.f; }
    if (l15 < 4) {
        const float* w2 = w_ih2 + l15 * H1_D + k0;
        #pragma unroll
        for (int kk = 0; kk < 16; ++kk) B2[kk] = *(const v2f*)(w2 + 4 * kk);
        bias2 = b_ih2[l15] + b_hh2[l15];
    }
    const float whh2_0 = w_hh2[0], whh2_1 = w_hh2[1];
    const float whh2_2 = w_hh2[2], whh2_3 = w_hh2[3];
    const float fcw = fc_w[0], fcb = fc_b[0];

    // per-lane x pointer: A-frag row m = l15 (batch), element ((b*T+t)*16 + 4*kk + k0)
    const float* xbase = x + (size_t)(b0 + l15) * T_LEN * IN_D + k0;

    // recurrent state: c1 as a C-fragment (lane l, reg r -> m = r + 8*lhalf, h = 16w + l15)
    v8f c1;
    #pragma unroll
    for (int r = 0; r < 8; ++r) c1[r] = 0.f;
    float h2 = 0.f, c2 = 0.f;                 // layer-2 scalar state (wave0 lanes 0-15)

    const int aoff = l15 * LDH + k0;          // A-frag LDS base (row m = l15)

    __syncthreads();

    for (int t = 0; t < T_LEN; ++t) {
        const int rp = t & 1, wp = rp ^ 1;

        // ---- A fragments: h_{t-1} from LDS, x_t from global ----
        v2f Ah[16], Axv[4];
        const float* hb = &hbuf[rp][aoff];
        #pragma unroll
        for (int kk = 0; kk < 16; ++kk) Ah[kk]  = *(const v2f*)(hb + 4 * kk);
        const float* xp = xbase + t * IN_D;
        #pragma unroll
        for (int kk = 0; kk < 4; ++kk)  Axv[kk] = *(const v2f*)(xp + 4 * kk);

        // ---- gates[16,256] tile GEMMs: 4 gate tiles x (16 + 4) K-chunks ----
        v8f acc[4];
        #pragma unroll
        for (int gi = 0; gi < 4; ++gi) {
            v8f c;
            #pragma unroll
            for (int r = 0; r < 8; ++r) c[r] = bias1[gi];
            #pragma unroll
            for (int kk = 0; kk < 16; ++kk)
                c = __builtin_amdgcn_wmma_f32_16x16x4_f32(
                        false, Ah[kk], false, Bh[gi][kk], (short)0, c, false, false);
            #pragma unroll
            for (int kk = 0; kk < 4; ++kk)
                c = __builtin_amdgcn_wmma_f32_16x16x4_f32(
                        false, Axv[kk], false, Bx[gi][kk], (short)0, c, false, false);
            acc[gi] = c;
        }

        // ---- LSTM cell (lane-local on C layout), write h_t slice to hbuf[wp] ----
        float* ho = &hbuf[wp][(lhalf * 8) * LDH + wave * 16 + l15];
        #pragma unroll
        for (int r = 0; r < 8; ++r) {
            float iv = sigf(acc[0][r]);
            float fv = sigf(acc[1][r]);
            float gv = tanhf(acc[2][r]);
            float ov = sigf(acc[3][r]);
            float cn = fv * c1[r] + iv * gv;
            c1[r] = cn;
            ho[r * LDH] = ov * tanhf(cn);
        }
        __syncthreads();   // h_t visible to all waves (and to wave0's layer 2)

        // ---- layer 2 (H=1) + FC on wave 0 ----
        if (wave == 0) {
            v8f c2acc;
            #pragma unroll
            for (int r = 0; r < 8; ++r) c2acc[r] = bias2;
            const float* hb2 = &hbuf[wp][aoff];
            #pragma unroll
            for (int kk = 0; kk < 16; ++kk) {
                v2f a2 = *(const v2f*)(hb2 + 4 * kk);
                c2acc = __builtin_amdgcn_wmma_f32_16x16x4_f32(
                            false, a2, false, B2[kk], (short)0, c2acc, false, false);
            }
            // scatter the 4 valid gate columns, then gather per batch row (intra-wave)
            if (l15 < 4) {
                #pragma unroll
                for (int r = 0; r < 8; ++r)
                    g2buf[(r + lhalf * 8) * 4 + l15] = c2acc[r];
            }
            if (lane < 16) {
                float gi2 = g2buf[lane * 4 + 0] + h2 * whh2_0;
                float gf2 = g2buf[lane * 4 + 1] + h2 * whh2_1;
                float gg2 = g2buf[lane * 4 + 2] + h2 * whh2_2;
                float go2 = g2buf[lane * 4 + 3] + h2 * whh2_3;
                float iv = sigf(gi2), fv = sigf(gf2);
                float gv = tanhf(gg2), ov = sigf(go2);
                c2 = fv * c2 + iv * gv;
                h2 = ov * tanhf(c2);
                out[(size_t)(b0 + lane) * T_LEN + t] = h2 * fcw + fcb;
            }
        }
    }
}

extern "C" void kernel_launch(void* const* d_in, const int* in_sizes, int n_in,
                              void* d_out, int out_size, void* d_ws, size_t ws_size,
                              hipStream_t stream) {
    (void)in_sizes; (void)n_in; (void)out_size; (void)d_ws; (void)ws_size;
    const float* x     = (const float*)d_in[0];
    const float* w_ih1 = (const float*)d_in[1];
    const float* w_hh1 = (const float*)d_in[2];
    const float* b_ih1 = (const float*)d_in[3];
    const float* b_hh1 = (const float*)d_in[4];
    const float* w_ih2 = (const float*)d_in[5];
    const float* w_hh2 = (const float*)d_in[6];
    const float* b_ih2 = (const float*)d_in[7];
    const float* b_hh2 = (const float*)d_in[8];
    const float* fc_w  = (const float*)d_in[9];
    const float* fc_b  = (const float*)d_in[10];
    float* out = (float*)d_out;

    dim3 grid(B_TOT / 16), block(128);
    lstm2_fused_kernel<<<grid, block, 0, stream>>>(
        x, w_ih1, w_hh1, b_ih1, b_hh1,
        w_ih2, w_hh2, b_ih2, b_hh2, fc_w, fc_b, out);
}